// NCF_Hyper_90357521973958
// MI455X (gfx1250) — compile-verified
//
#include <hip/hip_runtime.h>
#include <hip/hip_bf16.h>

typedef __attribute__((ext_vector_type(2))) float v2f;
typedef __attribute__((ext_vector_type(8))) float v8f;

#define N_CHUNKS   300
#define IN_DIM     96
#define HIDDEN     100
#define N_ROWS_WS  110000   // 11 * 10000
#define N_TILES_N  6875     // 110000 / 16
#define K_STEPS    25       // 100 / 4

// ---------------------------------------------------------------------------
// Kernel 1: hypernetwork trunk. One block per chunk.
//   p  = preference @ pref_emb                (32)
//   x  = [p | chunk_emb[c]]                   (96)
//   h1 = relu(x @ W1^T + b1)                  (100)
//   h2 = relu(h1 @ W2^T + b2)                 (100)
//   rep[c] = h2 @ W3^T + b3                   (100)
// ---------------------------------------------------------------------------
__global__ __launch_bounds__(128) void ncf_trunk_kernel(
    const float* __restrict__ pref,       // (2,)
    const float* __restrict__ pref_emb,   // (2,32)
    const float* __restrict__ chunk_emb,  // (300,64)
    const float* __restrict__ W1, const float* __restrict__ b1,   // (100,96),(100,)
    const float* __restrict__ W2, const float* __restrict__ b2,   // (100,100),(100,)
    const float* __restrict__ W3, const float* __restrict__ b3,   // (100,100),(100,)
    float* __restrict__ rep)              // (300,100) scratch
{
    __shared__ float x[IN_DIM];
    __shared__ float h1[HIDDEN];
    __shared__ float h2[HIDDEN];

    const int c = blockIdx.x;
    const int t = threadIdx.x;

    if (t < 32) x[t] = pref[0] * pref_emb[t] + pref[1] * pref_emb[32 + t];
    if (t < 64) x[32 + t] = chunk_emb[c * 64 + t];
    __syncthreads();

    if (t < HIDDEN) {
        float acc = b1[t];
        const float* w = W1 + t * IN_DIM;
        #pragma unroll 8
        for (int i = 0; i < IN_DIM; ++i) acc = fmaf(x[i], w[i], acc);
        h1[t] = fmaxf(acc, 0.0f);
    }
    __syncthreads();

    if (t < HIDDEN) {
        float acc = b2[t];
        const float* w = W2 + t * HIDDEN;
        #pragma unroll 10
        for (int i = 0; i < HIDDEN; ++i) acc = fmaf(h1[i], w[i], acc);
        h2[t] = fmaxf(acc, 0.0f);
    }
    __syncthreads();

    if (t < HIDDEN) {
        float acc = b3[t];
        const float* w = W3 + t * HIDDEN;
        #pragma unroll 10
        for (int i = 0; i < HIDDEN; ++i) acc = fmaf(h2[i], w[i], acc);
        rep[c * HIDDEN + t] = acc;
    }
}

// ---------------------------------------------------------------------------
// Kernel 2: OUT[c, m] = sum_k rep[c,k] * WS[m,k], WS = ws viewed (110000,100).
// One wave per 16x16 tile, V_WMMA_F32_16X16X4_F32, 25 K-steps.
//
// fp32 WMMA VGPR layouts (ISA 7.12.2):
//   A 16x4 : lanes 0-15 -> M=lane, {v0,v1} = K{0,1}; lanes 16-31 -> K{2,3}
//   B 4x16 : lanes 0-15 -> N=lane, {v0,v1} = K{0,1}; lanes 16-31 -> K{2,3}
//   C/D    : vgpr r, lanes 0-15 -> (M=r, N=lane); lanes 16-31 -> (M=r+8, N=lane-16)
// ---------------------------------------------------------------------------
__global__ __launch_bounds__(160) void ncf_gen_wmma_kernel(
    const float* __restrict__ rep,   // (300,100)
    const float* __restrict__ ws,    // (110000,100) row-major
    float* __restrict__ out,         // first out_size elements of (c*110000 + m)
    int out_size)
{
    const int wave   = (int)(threadIdx.x >> 5);          // 0..4, uniform in wave
    const int lane   = (int)(threadIdx.x & 31);
    const int lane16 = lane & 15;
    const int half   = lane >> 4;                        // 0: K{0,1}, 1: K{2,3}

    const int ntile = (int)blockIdx.x * 5 + wave;        // 0..6874, exact cover
    const int n0    = ntile * 16;
    const int c0    = (int)blockIdx.y * 16;              // 0,16,32,48,64

    const float* __restrict__ arow = rep + (c0 + lane16) * HIDDEN + 2 * half;
    const float* __restrict__ brow = ws  + (size_t)(n0 + lane16) * HIDDEN + 2 * half;

    v8f acc = {};
    #pragma unroll
    for (int s = 0; s < K_STEPS; ++s) {
        v2f a = *(const v2f*)(arow + 4 * s);   // rep[c0+lane16][4s+2h .. +1], 8B aligned
        v2f b = *(const v2f*)(brow + 4 * s);   // WS [n0+lane16][4s+2h .. +1], 8B aligned
        acc = __builtin_amdgcn_wmma_f32_16x16x4_f32(
            /*neg_a=*/false, a, /*neg_b=*/false, b,
            /*c_mod=*/(short)0, acc, /*reuse_a=*/false, /*reuse_b=*/false);
    }

    const long n = (long)(n0 + lane16);
    #pragma unroll
    for (int r = 0; r < 8; ++r) {
        const long c   = (long)(c0 + r + 8 * half);
        const long idx = c * (long)N_ROWS_WS + n;        // chunk-major flatten
        if (idx < (long)out_size) out[idx] = acc[r];
    }
}

extern "C" void kernel_launch(void* const* d_in, const int* in_sizes, int n_in,
                              void* d_out, int out_size, void* d_ws, size_t ws_size,
                              hipStream_t stream) {
    const float* pref      = (const float*)d_in[0];
    const float* pref_emb  = (const float*)d_in[1];
    const float* chunk_emb = (const float*)d_in[2];
    const float* W1 = (const float*)d_in[3];
    const float* b1 = (const float*)d_in[4];
    const float* W2 = (const float*)d_in[5];
    const float* b2 = (const float*)d_in[6];
    const float* W3 = (const float*)d_in[7];
    const float* b3 = (const float*)d_in[8];
    const float* ws = (const float*)d_in[9];

    float* out = (float*)d_out;
    float* rep = (float*)d_ws;   // 300*100 floats = 120 KB scratch

    ncf_trunk_kernel<<<dim3(N_CHUNKS), dim3(128), 0, stream>>>(
        pref, pref_emb, chunk_emb, W1, b1, W2, b2, W3, b3, rep);

    // N tiles: 6875 = 1375 blocks x 5 waves (exact, no tail -> EXEC all-1s).
    // M tiles: 5 (c = 0..79; writes masked to idx < out_size, i.e. c <= 64).
    ncf_gen_wmma_kernel<<<dim3(1375, 5), dim3(160), 0, stream>>>(
        rep, ws, out, out_size);
}